// Encoder_10350871183497
// MI455X (gfx1250) — compile-verified
//
#include <hip/hip_runtime.h>

namespace {

constexpr int NN   = 100000;   // nodes
constexpr int NE   = 1600000;  // edges
constexpr int FIN  = 128;
constexpr int FHID = 128;
constexpr int FOUT = 64;

typedef __attribute__((ext_vector_type(2))) float v2f;
typedef __attribute__((ext_vector_type(8))) float v8f;

// ------------------------------------------------------------------ utils
__global__ void k_zero(float* __restrict__ p, long n) {
  long i = (long)blockIdx.x * blockDim.x + threadIdx.x;
  long s = (long)gridDim.x * blockDim.x;
  for (; i < n; i += s) p[i] = 0.0f;
}

__global__ void k_deg_init(float* __restrict__ deg, int n) {
  int i = blockIdx.x * blockDim.x + threadIdx.x;
  if (i < n) deg[i] = 1.0f;               // self-loop contributes 1
}

__global__ void k_deg_count(const int* __restrict__ dst, float* __restrict__ deg, int e) {
  int i = blockIdx.x * blockDim.x + threadIdx.x;
  if (i < e) atomicAdd(deg + dst[i], 1.0f);
}

__global__ void k_dinv(float* __restrict__ deg, int n) {
  int i = blockIdx.x * blockDim.x + threadIdx.x;
  if (i < n) deg[i] = rsqrtf(deg[i]);     // deg >= 1 always
}

// ------------------------------------------------------------- WMMA GEMM
// C[M x NC] = A[M x K] * B[K x NC], fp32, one wave per 16x16 output tile.
// Block = (NC/16) waves; block b covers row tile b across all column tiles.
template <int K, int NC>
__global__ __launch_bounds__(256) void k_gemm_wmma(const float* __restrict__ A,
                                                   const float* __restrict__ B,
                                                   float* __restrict__ C) {
  const int lane = threadIdx.x & 31;
  const int wv   = threadIdx.x >> 5;      // column tile index
  const int l16  = lane & 15;
  const int half = lane >> 4;             // 0: K={0,1} / M=v ; 1: K={2,3} / M=v+8
  const int row  = blockIdx.x * 16 + l16;
  const int col  = wv * 16 + l16;

  const float* arow = A + (size_t)row * K;
  v8f acc = {};
#pragma unroll 4
  for (int k0 = 0; k0 < K; k0 += 4) {
    const int ka = k0 + half * 2;
    v2f a, b;
    a.x = arow[ka];
    a.y = arow[ka + 1];
    b.x = B[(size_t)ka * NC + col];
    b.y = B[(size_t)(ka + 1) * NC + col];
    acc = __builtin_amdgcn_wmma_f32_16x16x4_f32(false, a, false, b,
                                                (short)0, acc, false, false);
  }
  float* crow = C + (size_t)(blockIdx.x * 16 + half * 8) * NC + col;
#pragma unroll
  for (int v = 0; v < 8; v++) crow[(size_t)v * NC] = acc[v];
}

// --------------------------------------------------------- edge scatter
// One 32-lane wave per edge; lane covers features f = lane + 32*c (coalesced
// 128B gathers + atomics; h and agg are L2-resident at 51MB << 192MB L2).
template <int F>
__global__ void k_scatter(const float* __restrict__ h, const int* __restrict__ src,
                          const int* __restrict__ dst, const float* __restrict__ dinv,
                          float* __restrict__ agg, int e) {
  long gid  = (long)blockIdx.x * blockDim.x + threadIdx.x;
  int  edge = (int)(gid >> 5);
  int  lane = (int)(gid & 31);
  if (edge >= e) return;
  int   s = src[edge], d = dst[edge];
  float w = dinv[s] * dinv[d];
  const float* hs = h + (size_t)s * F;
  float*       ad = agg + (size_t)d * F;
#pragma unroll
  for (int c = 0; c < F / 32; c++) {
    int f = lane + 32 * c;
    atomicAdd(ad + f, hs[f] * w);
  }
}

// -------------------------------------------------- JAX Threefry-2x32-20
__device__ __forceinline__ unsigned rotl32(unsigned x, int r) {
  return (x << r) | (x >> (32 - r));
}

__device__ unsigned threefry2x32(unsigned k0, unsigned k1,
                                 unsigned c0, unsigned c1, int pick) {
  const unsigned ks2 = 0x1BD11BDAu ^ k0 ^ k1;
  unsigned x0 = c0 + k0;
  unsigned x1 = c1 + k1;
#define TF_R(r) { x0 += x1; x1 = rotl32(x1, r); x1 ^= x0; }
  TF_R(13) TF_R(15) TF_R(26) TF_R(6)
  x0 += k1;  x1 += ks2 + 1u;
  TF_R(17) TF_R(29) TF_R(16) TF_R(24)
  x0 += ks2; x1 += k0 + 2u;
  TF_R(13) TF_R(15) TF_R(26) TF_R(6)
  x0 += k0;  x1 += k1 + 3u;
  TF_R(17) TF_R(29) TF_R(16) TF_R(24)
  x0 += k1;  x1 += ks2 + 4u;
  TF_R(13) TF_R(15) TF_R(26) TF_R(6)
  x0 += ks2; x1 += k0 + 5u;
#undef TF_R
  return pick ? x1 : x0;
}

// self-loop + bias + relu + dropout(key=42, p=0.5); h2 may alias agg
__global__ void k_epilogue1(const float* __restrict__ agg, const float* __restrict__ h1,
                            const float* __restrict__ dinv, const float* __restrict__ b1,
                            float* __restrict__ h2) {
  const long total = (long)NN * FHID;
  long j = (long)blockIdx.x * blockDim.x + threadIdx.x;
  if (j >= total) return;
  int   i  = (int)(j >> 7);         // FHID = 128
  int   f  = (int)(j & 127);
  float di = dinv[i];
  float v  = agg[j] + di * di * h1[j] + b1[f];
  v = fmaxf(v, 0.0f);
  // JAX random_bits over flat (N*HID): halves of the counter iota
  const long h = total >> 1;
  unsigned bits = (j < h)
      ? threefry2x32(0u, 42u, (unsigned)j, (unsigned)(j + h), 0)
      : threefry2x32(0u, 42u, (unsigned)(j - h), (unsigned)j, 1);
  // keep = uniform(bits) < 0.5  <=>  top bit clear;  kept values scaled by 1/(1-p)=2
  h2[j] = (bits & 0x80000000u) ? 0.0f : v * 2.0f;
}

__global__ void k_epilogue2(const float* __restrict__ h3, const float* __restrict__ dinv,
                            const float* __restrict__ b2, float* __restrict__ out) {
  const long total = (long)NN * FOUT;
  long j = (long)blockIdx.x * blockDim.x + threadIdx.x;
  if (j >= total) return;
  int   i  = (int)(j >> 6);         // FOUT = 64
  int   f  = (int)(j & 63);
  float di = dinv[i];
  out[j] += di * di * h3[j] + b2[f];
}

}  // namespace

extern "C" void kernel_launch(void* const* d_in, const int* in_sizes, int n_in,
                              void* d_out, int out_size, void* d_ws, size_t ws_size,
                              hipStream_t stream) {
  (void)in_sizes; (void)n_in; (void)out_size; (void)ws_size;

  const float* x  = (const float*)d_in[0];
  const int*   ei = (const int*)d_in[1];
  const float* W1 = (const float*)d_in[2];
  const float* b1 = (const float*)d_in[3];
  const float* W2 = (const float*)d_in[4];
  const float* b2 = (const float*)d_in[5];
  const int* src = ei;            // edge_index[0]
  const int* dst = ei + NE;       // edge_index[1]
  float* out = (float*)d_out;

  // workspace layout (floats): h1 | agg(=h2) | dinv ; h3 reuses h1's region
  float* h1   = (float*)d_ws;                         // NN*FHID
  float* agg  = h1  + (size_t)NN * FHID;              // NN*FHID
  float* dinv = agg + (size_t)NN * FHID;              // NN
  float* h3   = h1;                                   // NN*FOUT (h1 dead by then)

  // ---- degrees -> dinv (shared by both layers)
  k_deg_init <<<(NN + 255) / 256, 256, 0, stream>>>(dinv, NN);
  k_deg_count<<<(NE + 255) / 256, 256, 0, stream>>>(dst, dinv, NE);
  k_dinv     <<<(NN + 255) / 256, 256, 0, stream>>>(dinv, NN);

  // ---- layer 1: h1 = x @ W1
  k_gemm_wmma<FIN, FHID><<<NN / 16, (FHID / 16) * 32, 0, stream>>>(x, W1, h1);
  k_zero<<<2048, 256, 0, stream>>>(agg, (long)NN * FHID);
  {
    long t = (long)NE * 32;
    k_scatter<FHID><<<(unsigned)((t + 255) / 256), 256, 0, stream>>>(h1, src, dst, dinv, agg, NE);
  }
  k_epilogue1<<<(unsigned)(((long)NN * FHID + 255) / 256), 256, 0, stream>>>(agg, h1, dinv, b1, agg);

  // ---- layer 2: h3 = h2 @ W2 ; out = scatter + self-loop + bias
  k_gemm_wmma<FHID, FOUT><<<NN / 16, (FOUT / 16) * 32, 0, stream>>>(agg, W2, h3);
  k_zero<<<2048, 256, 0, stream>>>(out, (long)NN * FOUT);
  {
    long t = (long)NE * 32;
    k_scatter<FOUT><<<(unsigned)((t + 255) / 256), 256, 0, stream>>>(h3, src, dst, dinv, out, NE);
  }
  k_epilogue2<<<(unsigned)(((long)NN * FOUT + 255) / 256), 256, 0, stream>>>(h3, dinv, b2, out);
}